// SwirlEffect_14199161880931
// MI455X (gfx1250) — compile-verified
//
#include <hip/hip_runtime.h>

// SwirlEffect: bilinear backward warp of (8,512,512,16) f32 with an analytic
// flow field scaled by frame_value. Memory-bound (AI ~0.8 FLOP/byte).
// gfx1250 shape: coalesced b128 gathers (L2-resident input), NT streaming
// b128 stores, and hardware TRANS ops (v_sin/v_cos/v_log) for the flow math
// since all arguments are in the fast range (|x| < 18) — no libm slow paths.

typedef __attribute__((ext_vector_type(4))) float float4v;

#define HH 512
#define WW 512
#define CC 16
#define BB 8

__global__ __launch_bounds__(256) void swirl_warp_kernel(
    const float* __restrict__ x,      // (8,512,512,16)
    const float* __restrict__ frame,  // (1,)
    float* __restrict__ out)          // (8,512,512,16)
{
    // lane layout: tid = pixel*4 + channel_quad
    const int tid = blockIdx.x * blockDim.x + threadIdx.x;
    const int cq  = tid & 3;          // which float4 of the 16 channels
    const int p   = tid >> 2;         // pixel index 0..512*512-1
    const int px  = p & (WW - 1);
    const int py  = p >> 9;

    const float fv = frame[0];        // uniform -> scalar load

    // ---- displacement field (matches _displacement(coeff=0.4, log_scale=2.0)) ----
    const float PIf = 3.14159265358979323846f;
    // y_flow = cos(pi * r[py]), r = arange(-1, 1, 1/256); |arg| <= pi -> hw cos
    const float ysv = -1.0f + (float)py * (1.0f / 256.0f);
    const float y_flow = __cosf(ysv * PIf);                 // v_cos_f32
    // xr = arange(-10pi, 10pi, 20pi/512); X = xr[px], Y = xr[py]
    const float step  = (float)(2.0 * 3.14159265358979323846 * 10.0 / 512.0);
    const float start = (float)(-3.14159265358979323846 * 10.0);
    const float Xc = start + (float)px * step;
    const float Yc = start + (float)py * step;
    const float wave = __builtin_sqrtf(Xc * Xc + Yc * Yc);  // v_sqrt_f32
    // sin arg <= 17.8, log arg in [~2e-5, 89] -> hw sin / hw log
    const float x_flow = __sinf(0.4f * wave) *
                         __logf(fmaxf(2.0f * wave, 1e-12f)); // v_sin_f32, v_log_f32

    // ---- backward-warp source coordinates + bilinear weights ----
    const float qy = (float)py - fv * y_flow;
    const float qx = (float)px - fv * x_flow;
    const float fy = fminf(fmaxf(__builtin_floorf(qy), 0.0f), (float)(HH - 2));
    const float fx = fminf(fmaxf(__builtin_floorf(qx), 0.0f), (float)(WW - 2));
    const float ay = fminf(fmaxf(qy - fy, 0.0f), 1.0f);
    const float ax = fminf(fmaxf(qx - fx, 0.0f), 1.0f);
    const int iy0 = (int)fy;
    const int ix0 = (int)fx;

    // flat offsets (floats)
    const long rowBase0 = ((long)iy0 * WW + ix0) * CC + (long)cq * 4;  // (iy0, ix0)
    const long rowBase1 = rowBase0 + (long)WW * CC;                    // (iy0+1, ix0)
    const long outBase  = (((long)py * WW) + px) * CC + (long)cq * 4;
    const long imgStride = (long)HH * WW * CC;                         // one batch slice

#pragma unroll
    for (int b = 0; b < BB; ++b) {
        const float* img = x + (long)b * imgStride;
        // 4 taps x float4 channels -> global_load_b128, coalesced across lanes
        const float4v tl = *(const float4v*)(img + rowBase0);
        const float4v tr = *(const float4v*)(img + rowBase0 + CC);
        const float4v bl = *(const float4v*)(img + rowBase1);
        const float4v br = *(const float4v*)(img + rowBase1 + CC);

        const float4v top = tl + ax * (tr - tl);
        const float4v bot = bl + ax * (br - bl);
        const float4v res = top + ay * (bot - top);

        // streaming write-once output: NT store keeps L2 for the gather reuse
        __builtin_nontemporal_store(res, (float4v*)(out + (long)b * imgStride + outBase));
    }
}

extern "C" void kernel_launch(void* const* d_in, const int* in_sizes, int n_in,
                              void* d_out, int out_size, void* d_ws, size_t ws_size,
                              hipStream_t stream) {
    (void)in_sizes; (void)n_in; (void)out_size; (void)d_ws; (void)ws_size;
    const float* x     = (const float*)d_in[0];   // (8,512,512,16) f32
    const float* frame = (const float*)d_in[1];   // (1,) f32
    float* out = (float*)d_out;

    // one thread per (pixel, channel-quad): 512*512*4 threads
    const int threads = HH * WW * 4;
    const int block = 256;
    const int grid = threads / block;             // 4096 blocks
    swirl_warp_kernel<<<grid, block, 0, stream>>>(x, frame, out);
}